// GinEncoderGraph_27358941675990
// MI455X (gfx1250) — compile-verified
//
#include <hip/hip_runtime.h>

typedef __attribute__((ext_vector_type(2))) float v2f;
typedef __attribute__((ext_vector_type(4))) float v4f;
typedef __attribute__((ext_vector_type(8))) float v8f;

#define N_NODES  50000
#define N_EDGES  800000
#define N_GRAPHS 256
#define N_HID    256
#define BN_EPS   1e-5f

// ---------------- row copy: buf[n][0..d) = H[n*ldH + 0..d) ----------------
__global__ void k_copy_rows(const float* __restrict__ H, int ldH, int d,
                            float* __restrict__ out, int nRows) {
  int chunks = d >> 2;
  long gid = (long)blockIdx.x * blockDim.x + threadIdx.x;
  long total = (long)nRows * chunks;
  if (gid >= total) return;
  int row = (int)(gid / chunks);
  int c4  = (int)(gid - (long)row * chunks) << 2;
  *(v4f*)(out + (long)row * d + c4) = *(const v4f*)(H + (long)row * ldH + c4);
}

// ---------------- edge scatter: out[dst] += H[src] ----------------
__global__ void k_edge_scatter(const float* __restrict__ H, int ldH, int d,
                               const int* __restrict__ src, const int* __restrict__ dst,
                               float* __restrict__ out, int nEdges) {
  int chunks = d >> 2;
  long gid = (long)blockIdx.x * blockDim.x + threadIdx.x;
  long total = (long)nEdges * chunks;
  if (gid >= total) return;
  int e  = (int)(gid / chunks);
  int c4 = (int)(gid - (long)e * chunks) << 2;
  int s = src[e], t = dst[e];
  v4f v = *(const v4f*)(H + (long)s * ldH + c4);
  float* o = out + (long)t * d + c4;
  atomicAdd(o + 0, v.x);
  atomicAdd(o + 1, v.y);
  atomicAdd(o + 2, v.z);
  atomicAdd(o + 3, v.w);
}

// ---------------- WMMA fp32 GEMM: C[M x 128-slice] = A[M x K] @ B[K x 256] ----------------
// Block: 256 threads = 8 waves. Block tile: 128(M) x 128(N). Wave: 16(M) x 128(N).
// Uses V_WMMA_F32_16X16X4_F32; A fragment per ISA layout (lanes 0-15: k,k+1 / 16-31: k+2,k+3).
#define MT 128
#define NT 128
#define KB 32
#define KPAD 36

__global__ void __launch_bounds__(256) k_wmma_gemm(
    const float* __restrict__ A, int lda,
    const float* __restrict__ B,      // K x 256 row-major
    float* __restrict__ C, int ldc,   // 128-col slice selected by blockIdx.y
    int M, int K, int relu) {
  __shared__ float As[MT][KPAD];      // row-major tile of A
  __shared__ float Bs[NT][KPAD];      // B tile transposed: Bs[n][k]

  const int tid  = threadIdx.x;
  const int lane = tid & 31;
  const int wave = tid >> 5;
  const int tileM = blockIdx.x * MT;
  const int tileN = blockIdx.y * NT;

  const v8f zero8 = {0.f,0.f,0.f,0.f,0.f,0.f,0.f,0.f};
  const v4f zero4 = {0.f,0.f,0.f,0.f};
  v8f acc[8];
#pragma unroll
  for (int j = 0; j < 8; ++j) acc[j] = zero8;

  // A-tile staging map: 2 threads per row, 16 cols each (4 x float4)
  const int arow  = tid >> 1;
  const int acol0 = (tid & 1) << 4;
  const int grow  = tileM + arow;
  const bool rowOK = (grow < M);

  // B-tile staging map: thread covers 1 k-row x 16 n-cols, stores transposed
  const int bk  = tid >> 3;          // 0..31
  const int bn0 = (tid & 7) << 4;    // 0..112

  const int klane = (lane >> 4) << 1;        // 0 or 2 (per 16x4 f32 A layout)
  const int mrow  = (wave << 4) + (lane & 15);
  const int nlane = lane & 15;

  for (int kb = 0; kb < K; kb += KB) {
#pragma unroll
    for (int i = 0; i < 4; ++i) {
      v4f v = rowOK ? *(const v4f*)(A + (long)grow * lda + kb + acol0 + i * 4) : zero4;
      *(v4f*)&As[arow][acol0 + i * 4] = v;
    }
    {
      const float* brow = B + (long)(kb + bk) * N_HID + tileN + bn0;
#pragma unroll
      for (int i = 0; i < 4; ++i) {
        v4f v = *(const v4f*)(brow + i * 4);
        Bs[bn0 + i * 4 + 0][bk] = v.x;
        Bs[bn0 + i * 4 + 1][bk] = v.y;
        Bs[bn0 + i * 4 + 2][bk] = v.z;
        Bs[bn0 + i * 4 + 3][bk] = v.w;
      }
    }
    __syncthreads();

#pragma unroll
    for (int ks = 0; ks < KB / 4; ++ks) {
      const int k0 = (ks << 2) + klane;
      v2f a = *(const v2f*)&As[mrow][k0];
#pragma unroll
      for (int j = 0; j < 8; ++j) {
        v2f b = *(const v2f*)&Bs[(j << 4) + nlane][k0];
        acc[j] = __builtin_amdgcn_wmma_f32_16x16x4_f32(
            /*neg_a=*/false, a, /*neg_b=*/false, b,
            /*c_mod=*/(short)0, acc[j], /*reuse_a=*/false, /*reuse_b=*/false);
      }
    }
    __syncthreads();
  }

  // C/D layout: lanes 0-15 -> M=v, lanes 16-31 -> M=8+v; N = lane&15
  const int mbase = tileM + (wave << 4) + ((lane >> 4) << 3);
  const int nbase = tileN + nlane;
#pragma unroll
  for (int j = 0; j < 8; ++j) {
#pragma unroll
    for (int v = 0; v < 8; ++v) {
      int row = mbase + v;
      if (row < M) {
        float val = acc[j][v];
        if (relu) val = fmaxf(val, 0.f);
        C[(long)row * ldc + nbase + (j << 4)] = val;
      }
    }
  }
}

// ---------------- BatchNorm: stats then apply ----------------
__global__ void k_zero(float* __restrict__ p, int n) {
  int i = blockIdx.x * blockDim.x + threadIdx.x;
  if (i < n) p[i] = 0.f;
}

__global__ void __launch_bounds__(256) k_bn_stats(const float* __restrict__ H, int M,
                                                  float* __restrict__ stats) {
  int col = threadIdx.x;             // 0..255
  int r0 = blockIdx.x * 64;
  float s = 0.f, s2 = 0.f;
  for (int r = 0; r < 64; ++r) {
    int row = r0 + r;
    if (row < M) {
      float v = H[(long)row * N_HID + col];
      s += v;
      s2 += v * v;
    }
  }
  atomicAdd(&stats[col], s);
  atomicAdd(&stats[N_HID + col], s2);
}

__global__ void k_bn_apply(const float* __restrict__ H, int M,
                           const float* __restrict__ stats,
                           const float* __restrict__ gamma, const float* __restrict__ beta,
                           float* __restrict__ out, int ldo) {
  long gid = (long)blockIdx.x * blockDim.x + threadIdx.x;
  long total = (long)M * N_HID;
  if (gid >= total) return;
  int row = (int)(gid >> 8);
  int col = (int)(gid & 255);
  float invM = 1.0f / (float)M;
  float mean = stats[col] * invM;
  float var  = stats[N_HID + col] * invM - mean * mean;
  float inv  = rsqrtf(var + BN_EPS);
  float v = H[gid];
  out[(long)row * ldo + col] = (v - mean) * inv * gamma[col] + beta[col];
}

// ---------------- global add pool ----------------
__global__ void k_pool(const float* __restrict__ H, int ldh,
                       const int* __restrict__ batch,
                       float* __restrict__ xpool, int nRows) {
  long gid = (long)blockIdx.x * blockDim.x + threadIdx.x;
  long total = (long)nRows * (N_HID >> 2);
  if (gid >= total) return;
  int row = (int)(gid >> 6);
  int c4  = (int)(gid & 63) << 2;
  v4f v = *(const v4f*)(H + (long)row * ldh + c4);
  float* o = xpool + (long)batch[row] * N_HID + c4;
  atomicAdd(o + 0, v.x);
  atomicAdd(o + 1, v.y);
  atomicAdd(o + 2, v.z);
  atomicAdd(o + 3, v.w);
}

extern "C" void kernel_launch(void* const* d_in, const int* in_sizes, int n_in,
                              void* d_out, int out_size, void* d_ws, size_t ws_size,
                              hipStream_t stream) {
  const float* x     = (const float*)d_in[0];
  const int*   edge  = (const int*)d_in[1];
  const int*   src   = edge;
  const int*   dst   = edge + N_EDGES;
  const int*   batch = (const int*)d_in[2];
  const float* W1_0 = (const float*)d_in[3];
  const float* W2_0 = (const float*)d_in[4];
  const float* g0   = (const float*)d_in[5];
  const float* b0   = (const float*)d_in[6];
  const float* W1_1 = (const float*)d_in[7];
  const float* W2_1 = (const float*)d_in[8];
  const float* g1   = (const float*)d_in[9];
  const float* b1   = (const float*)d_in[10];
  const float* W1_2 = (const float*)d_in[11];
  const float* W2_2 = (const float*)d_in[12];

  float* out   = (float*)d_out;
  float* xpool = out;                          // [256 x 256]
  float* xs    = out + N_GRAPHS * N_HID;       // [50000 x 768]  (concat slices at col 0/256/512)

  float* buf_sum = (float*)d_ws;                       // [50000 x 256]
  float* buf_mid = buf_sum + (long)N_NODES * N_HID;    // [50000 x 256]
  float* stats   = buf_mid + (long)N_NODES * N_HID;    // [512]

  auto cdiv = [](long a, long b) { return (unsigned)((a + b - 1) / b); };
  dim3 blk(256);
  dim3 gemmGrid(cdiv(N_NODES, MT), N_HID / NT);

  // ================= Layer 0 (in: x, d=128) =================
  k_copy_rows<<<cdiv((long)N_NODES * 32, 256), blk, 0, stream>>>(x, 128, 128, buf_sum, N_NODES);
  k_edge_scatter<<<cdiv((long)N_EDGES * 32, 256), blk, 0, stream>>>(x, 128, 128, src, dst, buf_sum, N_EDGES);
  k_wmma_gemm<<<gemmGrid, blk, 0, stream>>>(buf_sum, 128, W1_0, buf_mid, N_HID, N_NODES, 128, 1);
  k_wmma_gemm<<<gemmGrid, blk, 0, stream>>>(buf_mid, N_HID, W2_0, buf_sum, N_HID, N_NODES, N_HID, 1);
  k_zero<<<2, blk, 0, stream>>>(stats, 2 * N_HID);
  k_bn_stats<<<cdiv(N_NODES, 64), blk, 0, stream>>>(buf_sum, N_NODES, stats);
  k_bn_apply<<<cdiv((long)N_NODES * N_HID, 256), blk, 0, stream>>>(buf_sum, N_NODES, stats, g0, b0, xs + 0, 768);

  // ================= Layer 1 (in: xs[:,0:256], ld=768) =================
  k_copy_rows<<<cdiv((long)N_NODES * 64, 256), blk, 0, stream>>>(xs, 768, N_HID, buf_sum, N_NODES);
  k_edge_scatter<<<cdiv((long)N_EDGES * 64, 256), blk, 0, stream>>>(xs, 768, N_HID, src, dst, buf_sum, N_EDGES);
  k_wmma_gemm<<<gemmGrid, blk, 0, stream>>>(buf_sum, N_HID, W1_1, buf_mid, N_HID, N_NODES, N_HID, 1);
  k_wmma_gemm<<<gemmGrid, blk, 0, stream>>>(buf_mid, N_HID, W2_1, buf_sum, N_HID, N_NODES, N_HID, 1);
  k_zero<<<2, blk, 0, stream>>>(stats, 2 * N_HID);
  k_bn_stats<<<cdiv(N_NODES, 64), blk, 0, stream>>>(buf_sum, N_NODES, stats);
  k_bn_apply<<<cdiv((long)N_NODES * N_HID, 256), blk, 0, stream>>>(buf_sum, N_NODES, stats, g1, b1, xs + 256, 768);

  // ================= Layer 2 (in: xs[:,256:512]; no act/BN) =================
  k_copy_rows<<<cdiv((long)N_NODES * 64, 256), blk, 0, stream>>>(xs + 256, 768, N_HID, buf_sum, N_NODES);
  k_edge_scatter<<<cdiv((long)N_EDGES * 64, 256), blk, 0, stream>>>(xs + 256, 768, N_HID, src, dst, buf_sum, N_EDGES);
  k_wmma_gemm<<<gemmGrid, blk, 0, stream>>>(buf_sum, N_HID, W1_2, buf_mid, N_HID, N_NODES, N_HID, 1);
  k_wmma_gemm<<<gemmGrid, blk, 0, stream>>>(buf_mid, N_HID, W2_2, xs + 512, 768, N_NODES, N_HID, 0);

  // ================= Global add pool =================
  k_zero<<<cdiv(N_GRAPHS * N_HID, 256), blk, 0, stream>>>(xpool, N_GRAPHS * N_HID);
  k_pool<<<cdiv((long)N_NODES * 64, 256), blk, 0, stream>>>(xs + 512, 768, batch, xpool, N_NODES);
}